// TopK_83940840833382
// MI455X (gfx1250) — compile-verified
//
#include <hip/hip_runtime.h>

#define ROWS    8192
#define COLS    32768
#define TOPK    64
#define THREADS 1024
#define NWAVES  (THREADS / 32)
#define EPT     (COLS / THREADS)   // 32 elements per thread
#define V4PT    (EPT / 4)          // 8 vec4 per thread
#define EQCAP   128
#define WH_STRIDE 257              // pad so each wave's bin b lands in a different bank

typedef unsigned int u32x4 __attribute__((ext_vector_type(4)));
typedef int          i32x8 __attribute__((ext_vector_type(8)));
typedef int          i32x4 __attribute__((ext_vector_type(4)));
typedef float        f32x4 __attribute__((ext_vector_type(4)));

#ifdef __has_builtin
#if __has_builtin(__builtin_amdgcn_tensor_load_to_lds)
#define USE_TDM 1
#endif
#endif

// Order-preserving float -> uint key: unsigned compare == float compare.
__device__ __forceinline__ unsigned fkey(float f) {
    unsigned u = __float_as_uint(f);
    return (u & 0x80000000u) ? ~u : (u | 0x80000000u);
}

__global__ __launch_bounds__(THREADS)
void topk_relu_scatter_kernel(const float* __restrict__ x,
                              float* __restrict__ out_res,
                              float* __restrict__ out_msk,
                              float* __restrict__ out_idx) {
    __shared__ float    s_row[COLS];                 // 128 KB: full row in LDS
    __shared__ unsigned s_whist[NWAVES * WH_STRIDE]; // ~32.9 KB per-wave histograms (pass 1)
    __shared__ unsigned s_hist[256];
    __shared__ unsigned s_bm[COLS / 32];             // 4 KB membership bitmap
    __shared__ float    s_vals[TOPK];
    __shared__ int      s_idx[TOPK];
    __shared__ int      s_eq[EQCAP];
    __shared__ unsigned s_sel, s_above, s_gt, s_eqn;

    const int tid  = threadIdx.x;
    const int wave = tid >> 5;
    const int row  = blockIdx.x;
    const float* __restrict__ xr = x + (size_t)row * COLS;

    // ---- Stage the row into LDS ----
#if defined(USE_TDM)
    if (tid < 32) {  // wave 0 issues one TDM DMA for the whole 128KB row
        unsigned long long ga = (unsigned long long)(uintptr_t)xr;
        unsigned lds_off = (unsigned)(uintptr_t)&s_row[0];
        // D# group0: count=1 | lds_addr | global_addr[56:0] | type=2
        u32x4 g0 = { 1u,
                     lds_off,
                     (unsigned)(ga & 0xFFFFFFFFu),
                     (unsigned)((ga >> 32) & 0x1FFFFFFu) | 0x80000000u };
        // D# group1: data_size=4B; tensor_dim0=32768; tensor_dim1=1;
        //            tile_dim0=32768; tile_dim1=1; tensor_dim0_stride=32768
        i32x8 g1 = { 0x00020000,
                     (int)0x80000000,   // tensor_dim0[15:0] << 16
                     0x00010000,        // tensor_dim1[15:0] << 16
                     (int)0x80000000,   // tile_dim0 << 16
                     1,                 // tile_dim1
                     32768,             // tensor_dim0_stride (low 32)
                     0, 0 };
        i32x4 gz4 = { 0, 0, 0, 0 };
        i32x8 gz8 = { 0, 0, 0, 0, 0, 0, 0, 0 };
        __builtin_amdgcn_tensor_load_to_lds(g0, g1, gz4, gz4, gz8, 0);
        __builtin_amdgcn_s_wait_tensorcnt(0);
    }
#else
    {
        const f32x4* __restrict__ xr4 = (const f32x4*)xr;
        f32x4* s4 = (f32x4*)s_row;
        #pragma unroll
        for (int j = 0; j < V4PT; ++j) s4[tid + j * THREADS] = xr4[tid + j * THREADS];
    }
#endif

    // Zero per-wave histograms while the DMA is in flight (different LDS region).
    for (int j = tid; j < NWAVES * WH_STRIDE; j += THREADS) s_whist[j] = 0u;
    __syncthreads();

    // ---- Radix pass 1 (bits 31:24) with per-wave privatized histograms ----
    // Normal data concentrates ~29% of keys in one byte-bin; privatization keeps
    // the hot-bin atomic chains parallel across banks instead of one serial chain.
    {
        unsigned* wh = &s_whist[wave * WH_STRIDE];
        #pragma unroll
        for (int j = 0; j < EPT; ++j) {
            int i = tid + j * THREADS;
            unsigned kb = fkey(s_row[i]);
            atomicAdd(&wh[kb >> 24], 1u);
        }
    }
    __syncthreads();
    if (tid < 256) {
        unsigned s = 0;
        #pragma unroll
        for (int w = 0; w < NWAVES; ++w) s += s_whist[w * WH_STRIDE + tid];
        s_hist[tid] = s;
    }
    __syncthreads();

    unsigned prefix = 0u, pmask = 0u;
    unsigned remaining = TOPK;
    if (tid == 0) {
        unsigned cum = 0; int b = 255;
        for (; b > 0; --b) {
            unsigned c = s_hist[b];
            if (cum + c >= remaining) break;
            cum += c;
        }
        s_sel = (unsigned)b;
        s_above = cum;
    }
    __syncthreads();
    prefix |= s_sel << 24;
    pmask  |= 0xFFu << 24;
    remaining -= s_above;

    // ---- Radix passes 2-4: candidates shrink to O(100s), shared hist is fine ----
    for (int shift = 16; shift >= 0; shift -= 8) {
        if (tid < 256) s_hist[tid] = 0u;
        __syncthreads();
        #pragma unroll
        for (int j = 0; j < EPT; ++j) {
            int i = tid + j * THREADS;
            unsigned kb = fkey(s_row[i]);
            if ((kb & pmask) == prefix)
                atomicAdd(&s_hist[(kb >> shift) & 0xFFu], 1u);
        }
        __syncthreads();
        if (tid == 0) {
            unsigned cum = 0; int b = 255;
            for (; b > 0; --b) {
                unsigned c = s_hist[b];
                if (cum + c >= remaining) break;
                cum += c;
            }
            s_sel = (unsigned)b;
            s_above = cum;
        }
        __syncthreads();
        prefix |= s_sel << shift;
        pmask  |= 0xFFu << shift;
        remaining -= s_above;   // uniform across threads
    }
    const unsigned T = prefix;  // threshold key; `remaining` = #ties to take

    // ---- Gather strictly-greater elements and threshold ties ----
    if (tid == 0) { s_gt = 0u; s_eqn = 0u; }
    __syncthreads();
    #pragma unroll
    for (int j = 0; j < EPT; ++j) {
        int i = tid + j * THREADS;
        unsigned kb = fkey(s_row[i]);
        if (kb > T) {
            unsigned p = atomicAdd(&s_gt, 1u);
            if (p < TOPK) { s_vals[p] = s_row[i]; s_idx[p] = i; }
        } else if (kb == T) {
            unsigned p = atomicAdd(&s_eqn, 1u);
            if (p < EQCAP) s_eq[p] = i;
        }
    }
    __syncthreads();

    // ---- Thread 0: resolve ties (lowest index first, like jax top_k), sort 64 ----
    if (tid == 0) {
        int ngt  = (int)s_gt;          // == TOPK - remaining by construction
        int need = TOPK - ngt;
        int ne   = (int)s_eqn; if (ne > EQCAP) ne = EQCAP;
        for (int a = 1; a < ne; ++a) {  // sort tie indices ascending
            int v = s_eq[a]; int b = a - 1;
            while (b >= 0 && s_eq[b] > v) { s_eq[b + 1] = s_eq[b]; --b; }
            s_eq[b + 1] = v;
        }
        unsigned tb = (T & 0x80000000u) ? (T & 0x7FFFFFFFu) : ~T;
        float tv = __uint_as_float(tb);
        for (int a = 0; a < need && a < ne; ++a) { s_vals[ngt + a] = tv; s_idx[ngt + a] = s_eq[a]; }
        // insertion sort 64 entries: value desc, index asc
        for (int a = 1; a < TOPK; ++a) {
            float v = s_vals[a]; int ix = s_idx[a]; int b = a - 1;
            while (b >= 0 && (s_vals[b] < v || (s_vals[b] == v && s_idx[b] > ix))) {
                s_vals[b + 1] = s_vals[b]; s_idx[b + 1] = s_idx[b]; --b;
            }
            s_vals[b + 1] = v; s_idx[b + 1] = ix;
        }
    }

    // ---- Membership bitmap + idx output ----
    for (int j = tid; j < COLS / 32; j += THREADS) s_bm[j] = 0u;
    __syncthreads();
    if (tid < TOPK) {
        int i = s_idx[tid];
        atomicOr(&s_bm[i >> 5], 1u << (i & 31));
        out_idx[(size_t)row * TOPK + tid] = (float)i;
    }
    __syncthreads();

    // ---- Single coalesced streaming pass: dense result + mask (NT stores) ----
    float* __restrict__ res = out_res + (size_t)row * COLS;
    float* __restrict__ msk = out_msk + (size_t)row * COLS;
    #pragma unroll
    for (int j = 0; j < V4PT; ++j) {
        int i4 = tid + j * THREADS;
        int base = i4 * 4;
        f32x4 rv, mv;
        #pragma unroll
        for (int e = 0; e < 4; ++e) {
            int i = base + e;
            unsigned in = (s_bm[i >> 5] >> (i & 31)) & 1u;
            float v = s_row[i];
            rv[e] = in ? (v > 0.0f ? v : 0.0f) : 0.0f;
            mv[e] = in ? 1.0f : 0.0f;
        }
        __builtin_nontemporal_store(rv, (f32x4*)(res + base));
        __builtin_nontemporal_store(mv, (f32x4*)(msk + base));
    }
}

extern "C" void kernel_launch(void* const* d_in, const int* in_sizes, int n_in,
                              void* d_out, int out_size, void* d_ws, size_t ws_size,
                              hipStream_t stream) {
    (void)in_sizes; (void)n_in; (void)out_size; (void)d_ws; (void)ws_size;
    const float* x = (const float*)d_in[0];
    float* out = (float*)d_out;
    const size_t N = (size_t)ROWS * COLS;
    float* res  = out;           // result [8192,32768]
    float* msk  = out + N;       // mask   [8192,32768] (as 0/1 float)
    float* fidx = out + 2 * N;   // idx    [8192,64]    (as float)
    topk_relu_scatter_kernel<<<dim3(ROWS), dim3(THREADS), 0, stream>>>(x, res, msk, fidx);
}